// GNNLoss_36790689858125
// MI455X (gfx1250) — compile-verified
//
#include <hip/hip_runtime.h>

typedef __attribute__((ext_vector_type(2))) float v2f;
typedef __attribute__((ext_vector_type(8))) float v8f;
typedef __attribute__((ext_vector_type(4))) float f32x4;
typedef __attribute__((ext_vector_type(4))) int   i32x4;

#define ALPHA_C 0.25f

// XOR lane-swap via ds_swizzle_b32 immediate (group-of-32 mode):
// lane' = (lane & 0x1f) ^ mask  ->  imm = (mask << 10) | 0x1f
#define XOR_SWZ(x, mask)                                                    \
    __int_as_float(__builtin_amdgcn_ds_swizzle(__float_as_int(x),           \
                                               (((mask) << 10) | 0x1f)))

// Stable sigmoid-focal-loss contribution (gamma=2) + accuracy contribution.
// Shares one exp(-|x|) between BCE's log1p term and the sigmoid.
__device__ __forceinline__ void focal_accum(float x, float t,
                                            float& lossAcc, float& accAcc) {
    float ax  = __builtin_fabsf(x);
    float e   = __expf(-ax);                  // exp(-|x|) in (0, 1]
    float lg  = __logf(1.0f + e);             // log1p(exp(-|x|)), arg in (1, 2]
    float ce  = fmaxf(x, 0.0f) - x * t + lg;  // BCE-with-logits
    float inv = __builtin_amdgcn_rcpf(1.0f + e);
    float p   = (x >= 0.0f) ? inv : e * inv;  // sigmoid(x)
    float p_t = t * p + (1.0f - t) * (1.0f - p);
    float om  = 1.0f - p_t;
    float at  = t * ALPHA_C + (1.0f - t) * (1.0f - ALPHA_C);
    lossAcc   = __builtin_fmaf(at * ce, om * om, lossAcc);
    float pred = (x > 0.0f) ? 1.0f : 0.0f;    // sigmoid(x) > 0.5  <=>  x > 0
    accAcc    += (pred == t) ? 1.0f : 0.0f;
}

// Dual cross-lane sum via V_WMMA_F32_16X16X4_F32.
// A (16x4) selector: row0 = [1,0,1,0], row1 = [0,1,0,1], other rows 0.
// B (4x16) per-lane: b.x = loss partial, b.y = acc partial.
// => D[0][n] = loss(n)+loss(n+16)  in c[0], lanes 0..15
//    D[1][n] = acc(n) +acc(n+16)   in c[1], lanes 0..15
// Rows 8/9 (lanes 16..31 of c[0]/c[1]) are zero, so the WMMA performs the
// offset-16 fold; ds_swizzle XOR folds 8/4/2/1 finish the reduction.
__device__ __forceinline__ void wave_reduce_atomic(float lossAcc, float accAcc,
                                                   float* wsLoss, float* wsAcc) {
    const int lane = threadIdx.x & 31;
    v2f a;
    a.x = ((lane & 15) == 0) ? 1.0f : 0.0f;
    a.y = ((lane & 15) == 1) ? 1.0f : 0.0f;
    v2f b;
    b.x = lossAcc;
    b.y = accAcc;
    v8f c = {};
    c = __builtin_amdgcn_wmma_f32_16x16x4_f32(
            /*neg_a=*/false, a, /*neg_b=*/false, b,
            /*c_mod=*/(short)0, c, /*reuse_a=*/false, /*reuse_b=*/false);
    float l = c[0];
    float v = c[1];
    l += XOR_SWZ(l, 8);  v += XOR_SWZ(v, 8);
    l += XOR_SWZ(l, 4);  v += XOR_SWZ(v, 4);
    l += XOR_SWZ(l, 2);  v += XOR_SWZ(v, 2);
    l += XOR_SWZ(l, 1);  v += XOR_SWZ(v, 1);
    if (lane == 0) {
        atomicAdd(wsLoss, l);
        atomicAdd(wsAcc,  v);
    }
}

__device__ __forceinline__ float edge_target(int s, int d,
                                             const int* __restrict__ batch,
                                             const int* __restrict__ pinst) {
    int ps = pinst[s];              // regular (RT) loads: tables stay cached
    int pd = pinst[d];
    // (ps==pd && ps!=0) implies pd!=0
    bool same = (ps == pd) & (batch[s] == batch[d]) & (ps != 0);
    return same ? 1.0f : 0.0f;
}

__global__ void __launch_bounds__(256)
edge_kernel(const float* __restrict__ elog,
            const int*   __restrict__ eidx,   // [2, E] flattened
            const int*   __restrict__ batch,
            const int*   __restrict__ pinst,
            float* __restrict__ ws, unsigned int E) {
    const unsigned int tid   = blockIdx.x * blockDim.x + threadIdx.x;
    const unsigned int T     = gridDim.x * blockDim.x;
    const unsigned int nvec  = (E + 3u) >> 2;
    const unsigned int iters = (nvec + T - 1u) / T;
    const unsigned int end   = iters * T;       // uniform trip count for all lanes
    const bool vecOK = ((E & 3u) == 0u);        // dst-row b128 alignment

    float lossAcc = 0.0f, accAcc = 0.0f;
    for (unsigned int v = tid; v < end; v += T) {
        const unsigned int e0 = v << 2;
        if (vecOK && (e0 + 3u < E)) {
            // One-touch streams: non-temporal so gather tables stay in cache.
            const f32x4 x4 = __builtin_nontemporal_load((const f32x4*)(elog + e0));
            const i32x4 s4 = __builtin_nontemporal_load((const i32x4*)(eidx + e0));
            const i32x4 d4 = __builtin_nontemporal_load((const i32x4*)(eidx + E + e0));
#pragma unroll
            for (int j = 0; j < 4; ++j) {
                focal_accum(x4[j], edge_target(s4[j], d4[j], batch, pinst),
                            lossAcc, accAcc);
            }
        } else {
#pragma unroll
            for (int j = 0; j < 4; ++j) {
                const unsigned int e = e0 + j;
                if (e < E) {
                    focal_accum(elog[e],
                                edge_target(eidx[e], eidx[E + e], batch, pinst),
                                lossAcc, accAcc);
                }
            }
        }
    }
    // No divergent returns above: EXEC is all 1s for the WMMA.
    wave_reduce_atomic(lossAcc, accAcc, ws + 0, ws + 1);
}

__global__ void __launch_bounds__(256)
node_kernel(const float* __restrict__ nlog,
            const int*   __restrict__ pinst,
            float* __restrict__ ws, unsigned int N) {
    const unsigned int tid   = blockIdx.x * blockDim.x + threadIdx.x;
    const unsigned int T     = gridDim.x * blockDim.x;
    const unsigned int nvec  = (N + 3u) >> 2;
    const unsigned int iters = (nvec + T - 1u) / T;
    const unsigned int end   = iters * T;
    const bool vecOK = ((N & 3u) == 0u);

    float lossAcc = 0.0f, accAcc = 0.0f;
    for (unsigned int v = tid; v < end; v += T) {
        const unsigned int i0 = v << 2;
        if (vecOK && (i0 + 3u < N)) {
            const f32x4 x4 = __builtin_nontemporal_load((const f32x4*)(nlog + i0));
            const i32x4 p4 = __builtin_nontemporal_load((const i32x4*)(pinst + i0));
#pragma unroll
            for (int j = 0; j < 4; ++j) {
                focal_accum(x4[j], (p4[j] != 0) ? 1.0f : 0.0f, lossAcc, accAcc);
            }
        } else {
#pragma unroll
            for (int j = 0; j < 4; ++j) {
                const unsigned int i = i0 + j;
                if (i < N) {
                    focal_accum(nlog[i], (pinst[i] != 0) ? 1.0f : 0.0f,
                                lossAcc, accAcc);
                }
            }
        }
    }
    wave_reduce_atomic(lossAcc, accAcc, ws + 2, ws + 3);
}

__global__ void zero_ws_kernel(float* __restrict__ ws) {
    if (threadIdx.x < 4) ws[threadIdx.x] = 0.0f;
}

__global__ void finalize_kernel(const float* __restrict__ ws,
                                float* __restrict__ out,
                                float invE, float invN) {
    if (threadIdx.x == 0) {
        float el = ws[0] * invE;
        float ea = ws[1] * invE;
        float nl = ws[2] * invN;
        float na = ws[3] * invN;
        out[0] = el + nl;  // loss
        out[1] = el;       // edge_loss
        out[2] = nl;       // node_loss
        out[3] = ea;       // edge_acc
        out[4] = na;       // node_acc
    }
}

extern "C" void kernel_launch(void* const* d_in, const int* in_sizes, int n_in,
                              void* d_out, int out_size, void* d_ws, size_t ws_size,
                              hipStream_t stream) {
    const float* elog  = (const float*)d_in[0];   // edge_logits [E]
    const float* nlog  = (const float*)d_in[1];   // node_logits [N]
    const int*   batch = (const int*)  d_in[2];   // batch [N]
    const int*   pinst = (const int*)  d_in[3];   // point_instances [N]
    const int*   eidx  = (const int*)  d_in[4];   // edge_index [2, E]

    const unsigned int E = (unsigned int)in_sizes[0];
    const unsigned int N = (unsigned int)in_sizes[1];

    float* ws  = (float*)d_ws;
    float* out = (float*)d_out;

    zero_ws_kernel<<<1, 32, 0, stream>>>(ws);

    // Edge pass: 1024 blocks x 256 threads -> ~12 vec4 iterations/thread.
    edge_kernel<<<1024, 256, 0, stream>>>(elog, eidx, batch, pinst, ws, E);
    // Node pass: small (200k elements).
    node_kernel<<<128, 256, 0, stream>>>(nlog, pinst, ws, N);

    finalize_kernel<<<1, 32, 0, stream>>>(ws, out,
                                          1.0f / (float)E, 1.0f / (float)N);
}